// ttBloomMLP_6820408066235
// MI455X (gfx1250) — compile-verified
//
#include <hip/hip_runtime.h>
#include <hip/hip_bf16.h>

typedef float v2f __attribute__((ext_vector_type(2)));
typedef float v8f __attribute__((ext_vector_type(8)));

#define HID 64
#define DFF 256
#define WAVES 8
#define THREADS (WAVES * 32)

// Bloom tanh-GELU: 0.5*x*(1 + tanh(0.79788456*(x + 0.044715*x^3)))
__device__ __forceinline__ float bloom_gelu(float x) {
    float t = tanhf(0.79788456f * (x + 0.044715f * x * x * x));
    return 0.5f * x * (1.0f + t);
}

__global__ __launch_bounds__(THREADS, 1)
void bloom_mlp_wmma(const float* __restrict__ X,
                    const float* __restrict__ R,
                    const float* __restrict__ W1,   // [DFF][HID]
                    const float* __restrict__ b1,   // [DFF]
                    const float* __restrict__ W2,   // [HID][DFF]
                    const float* __restrict__ b2,   // [HID]
                    float* __restrict__ out,
                    int nTiles) {
    __shared__ float sW1[DFF * HID];          // 64 KB, row-major [f][d]
    __shared__ float sW2[HID * DFF];          // 64 KB, row-major [d][f]
    __shared__ float sB1[DFF];                // 1 KB
    __shared__ float sStage[WAVES][16 * 18];  // per-wave 16x16 bounce, padded stride 18

    const int tid = threadIdx.x;

    // ---- cooperative weight staging into LDS (float4 vectorized) ----
    {
        const float4* w1v = reinterpret_cast<const float4*>(W1);
        float4* s1v = reinterpret_cast<float4*>(sW1);
        for (int i = tid; i < (DFF * HID) / 4; i += THREADS) s1v[i] = w1v[i];
        const float4* w2v = reinterpret_cast<const float4*>(W2);
        float4* s2v = reinterpret_cast<float4*>(sW2);
        for (int i = tid; i < (HID * DFF) / 4; i += THREADS) s2v[i] = w2v[i];
        for (int i = tid; i < DFF; i += THREADS) sB1[i] = b1[i];
    }
    __syncthreads();

    const int lane  = tid & 31;
    const int wave  = tid >> 5;
    const int lrow  = lane & 15;   // M row (A/C) or N col (B/C)
    const int lhalf = lane >> 4;   // K-half selector
    float* stage = &sStage[wave][0];

    // b2 per output d-tile: C-layout element column = dd*16 + lrow
    float c2[4];
#pragma unroll
    for (int dd = 0; dd < 4; ++dd) c2[dd] = b2[dd * 16 + lrow];

    const int waveStride = gridDim.x * WAVES;
#pragma unroll 1
    for (int tile = blockIdx.x * WAVES + wave; tile < nTiles; tile += waveStride) {
        const size_t tok0 = (size_t)tile * 16;
        const float* Xt = X + tok0 * HID;

        // ---- load X tile as 16 A-fragments (16x4 f32 each) ----
        // A elem (m,k): lane holds m=lrow, k = kk*4 + 2*lhalf + {0,1}
        v2f a[16];
        {
            const float2* xrow =
                reinterpret_cast<const float2*>(Xt + (size_t)lrow * HID + lhalf * 2);
#pragma unroll
            for (int kk = 0; kk < 16; ++kk) {
                float2 t = xrow[kk * 2];
                a[kk].x = t.x; a[kk].y = t.y;
            }
        }

        // ---- output accumulators, seeded with b2 ----
        v8f acc[4];
#pragma unroll
        for (int dd = 0; dd < 4; ++dd) {
#pragma unroll
            for (int j = 0; j < 8; ++j) acc[dd][j] = c2[dd];
        }

        // ---- fused loop over 16 f-tiles of the hidden (DFF) dimension ----
#pragma unroll 1
        for (int ff = 0; ff < 16; ++ff) {
            // GEMM1 accumulator seeded with b1 (depends only on N = ff*16+lrow)
            float bias = sB1[ff * 16 + lrow];
            v8f h;
#pragma unroll
            for (int j = 0; j < 8; ++j) h[j] = bias;

            // B1 frag (k,n): n=lrow -> f = ff*16+lrow; k pair = W1[f][kk*4+2*lhalf+{0,1}]
            const float2* w1row = reinterpret_cast<const float2*>(
                &sW1[(ff * 16 + lrow) * HID + lhalf * 2]);
#pragma unroll
            for (int kk = 0; kk < 16; ++kk) {
                float2 t = w1row[kk * 2];
                v2f bfrag; bfrag.x = t.x; bfrag.y = t.y;
                h = __builtin_amdgcn_wmma_f32_16x16x4_f32(
                        false, a[kk], false, bfrag, (short)0, h, false, false);
            }

            // GELU in-register
#pragma unroll
            for (int j = 0; j < 8; ++j) h[j] = bloom_gelu(h[j]);

            // C-layout -> staging: elem j is [m = j+8*lhalf][n = lrow]
#pragma unroll
            for (int j = 0; j < 8; ++j)
                stage[(j + 8 * lhalf) * 18 + lrow] = h[j];
            __builtin_amdgcn_wave_barrier();  // keep DS store->load order at compile time

            // GEMM2: A-frag from staging, B-frag from sW2; accumulate 4 d-tiles
#pragma unroll
            for (int r2 = 0; r2 < 4; ++r2) {
                const float2* sp = reinterpret_cast<const float2*>(
                    &stage[lrow * 18 + r2 * 4 + lhalf * 2]);
                float2 gt = sp[0];
                v2f g; g.x = gt.x; g.y = gt.y;
                const int kbase = ff * 16 + r2 * 4 + lhalf * 2;
#pragma unroll
                for (int dd = 0; dd < 4; ++dd) {
                    const float2* wp = reinterpret_cast<const float2*>(
                        &sW2[(dd * 16 + lrow) * DFF + kbase]);
                    float2 wt = wp[0];
                    v2f bfrag; bfrag.x = wt.x; bfrag.y = wt.y;
                    acc[dd] = __builtin_amdgcn_wmma_f32_16x16x4_f32(
                                  false, g, false, bfrag, (short)0, acc[dd], false, false);
                }
            }
            __builtin_amdgcn_wave_barrier();  // protect staging reuse across ff iters
        }

        // ---- epilogue: + residual, store (C-layout: [m=j+8*lhalf][col=dd*16+lrow]) ----
        const float* Rt = R + tok0 * HID;
        float* Ot = out + tok0 * HID;
#pragma unroll
        for (int dd = 0; dd < 4; ++dd) {
#pragma unroll
            for (int j = 0; j < 8; ++j) {
                const int m = j + 8 * lhalf;
                const size_t idx = (size_t)m * HID + dd * 16 + lrow;
                Ot[idx] = acc[dd][j] + Rt[idx];
            }
        }
    }
}

extern "C" void kernel_launch(void* const* d_in, const int* in_sizes, int n_in,
                              void* d_out, int out_size, void* d_ws, size_t ws_size,
                              hipStream_t stream) {
    const float* X  = (const float*)d_in[0];
    const float* R  = (const float*)d_in[1];
    const float* W1 = (const float*)d_in[2];
    const float* b1 = (const float*)d_in[3];
    const float* W2 = (const float*)d_in[4];
    const float* b2 = (const float*)d_in[5];
    float* out = (float*)d_out;

    const int N = in_sizes[0] / HID;      // tokens
    const int nTiles = N / 16;            // 16-token tiles (N = 262144 -> 16384)

    int blocks = (nTiles + WAVES - 1) / WAVES;
    if (blocks > 1024) blocks = 1024;     // persistent-ish waves, grid-stride over tiles
    if (blocks < 1) blocks = 1;

    bloom_mlp_wmma<<<blocks, THREADS, 0, stream>>>(X, R, W1, b1, W2, b2, out, nTiles);
}